// Module_77446850281650
// MI455X (gfx1250) — compile-verified
//
#include <hip/hip_runtime.h>
#include <math.h>

#ifndef __has_builtin
#define __has_builtin(x) 0
#endif

typedef __attribute__((ext_vector_type(2))) float v2f;
typedef __attribute__((ext_vector_type(8))) float v8f;
typedef __attribute__((ext_vector_type(4))) int   v4i;

#define SIGMA_ 0.1f
#define B_ 32
#define H_ 8192
#define D_ 64
#define ROWS_PER_BLOCK 128            // 8 waves * 16 rows
#define NCHUNK (H_ / ROWS_PER_BLOCK)  // 64

// ---------------- block reduction helpers (256 threads) ----------------
__device__ __forceinline__ float block_reduce_sum(float v, float* red) {
  const int tid = threadIdx.x;
  red[tid] = v; __syncthreads();
  #pragma unroll
  for (int s = 128; s > 0; s >>= 1) {
    if (tid < s) red[tid] = red[tid] + red[tid + s];
    __syncthreads();
  }
  float r = red[0]; __syncthreads();
  return r;
}

__device__ __forceinline__ float block_reduce_max(float v, float* red) {
  const int tid = threadIdx.x;
  red[tid] = v; __syncthreads();
  #pragma unroll
  for (int s = 128; s > 0; s >>= 1) {
    if (tid < s) red[tid] = fmaxf(red[tid], red[tid + s]);
    __syncthreads();
  }
  float r = red[0]; __syncthreads();
  return r;
}

// ---------------- kernel 1: mu = (Q*w) . K  via WMMA f32 16x16x4 ----------------
// grid (NCHUNK, B), block 256. Each wave produces 16 rows of mu; each block 128.
__global__ __launch_bounds__(256)
void k1_mu(const float* __restrict__ Q, const float* __restrict__ K,
           const int* __restrict__ mask, const float* __restrict__ eps,
           const float* __restrict__ w_score,
           float* __restrict__ mu_out, float* __restrict__ z_ws)
{
  __shared__ float ldsQw[D_];
  __shared__ float ldsMu[ROWS_PER_BLOCK];

  const int b     = blockIdx.y;
  const int chunk = blockIdx.x;
  const int tid   = threadIdx.x;
  const int lane  = tid & 31;
  const int wave  = tid >> 5;

  if (tid < D_) ldsQw[tid] = Q[b * D_ + tid] * w_score[tid];
  __syncthreads();

#if __has_builtin(__builtin_amdgcn_wmma_f32_16x16x4_f32)
  // B operand (4x16 f32): column N=0 holds Qw[d0..d0+3], everything else zero.
  // Layout (mirrors documented 16x4 A layout): lane 0 -> (K=0,K=1), lane 16 -> (K=2,K=3).
  v2f bmat[16];
  #pragma unroll
  for (int c = 0; c < 16; ++c) {
    v2f bv; bv.x = 0.0f; bv.y = 0.0f;
    if (lane == 0)       { bv.x = ldsQw[4 * c + 0]; bv.y = ldsQw[4 * c + 1]; }
    else if (lane == 16) { bv.x = ldsQw[4 * c + 2]; bv.y = ldsQw[4 * c + 3]; }
    bmat[c] = bv;
  }

  const int hbase = chunk * ROWS_PER_BLOCK + wave * 16;
  // A operand (16x4 f32): lane l<16 holds row M=l cols (K=0,1); lane>=16 row M=l-16 cols (K=2,3)
  const float* krow = K + ((size_t)(b * H_ + hbase + (lane & 15))) * D_ + ((lane >> 4) * 2);

  v2f a[16];
  #pragma unroll
  for (int c = 0; c < 16; ++c)
    a[c] = *(const v2f*)(krow + 4 * c);

  v8f acc = {0.f, 0.f, 0.f, 0.f, 0.f, 0.f, 0.f, 0.f};
  #pragma unroll
  for (int c = 0; c < 16; ++c)
    acc = __builtin_amdgcn_wmma_f32_16x16x4_f32(false, a[c], false, bmat[c],
                                                (short)0, acc, false, false);

  // Column N=0 of the 16x16 f32 D tile: lane 0 holds M=0..7 in acc[0..7],
  // lane 16 holds M=8..15 in acc[0..7].
  if (lane == 0 || lane == 16) {
    const int base = wave * 16 + ((lane >> 4) << 3);
    #pragma unroll
    for (int v = 0; v < 8; ++v) ldsMu[base + v] = acc[v];
  }
#else
  // VALU fallback (no WMMA builtin available)
  if (tid < ROWS_PER_BLOCK) {
    const float* kr = K + ((size_t)(b * H_ + chunk * ROWS_PER_BLOCK + tid)) * D_;
    float s = 0.0f;
    #pragma unroll
    for (int dd = 0; dd < D_; ++dd) s += kr[dd] * ldsQw[dd];
    ldsMu[tid] = s;
  }
#endif
  __syncthreads();

  // Write mu to output and z = mask ? -inf : exp(mu + sigma*eps) to workspace.
  if (tid < ROWS_PER_BLOCK) {
    const int h = chunk * ROWS_PER_BLOCK + tid;
    const size_t idx = (size_t)b * H_ + h;
    const float m = ldsMu[tid];
    mu_out[idx] = m;
    const float zv = mask[idx] ? -__builtin_inff() : expf(m + SIGMA_ * eps[idx]);
    z_ws[idx] = zv;
  }
}

// ---------------- kernel 2: per-row simplex mix (sparsemax + softmax) ----------------
// grid (B), block 256. z row lives in LDS; Michelot iteration replaces the sort.
__global__ __launch_bounds__(256)
void k2_simplex(float* __restrict__ zw)  // in: z   out: weights (overwritten)
{
  __shared__ float ldsZ[H_];
  __shared__ float red[256];

  const int b = blockIdx.x;
  const int tid = threadIdx.x;
  float* zr = zw + (size_t)b * H_;

  for (int h = tid; h < H_; h += 256) ldsZ[h] = zr[h];
  __syncthreads();

  // softmax: max then sum of exp
  float lmax = -__builtin_inff();
  for (int h = tid; h < H_; h += 256) lmax = fmaxf(lmax, ldsZ[h]);
  const float zmax = block_reduce_max(lmax, red);

  float lsum = 0.0f;
  for (int h = tid; h < H_; h += 256) lsum += expf(ldsZ[h] - zmax);
  const float denom = block_reduce_sum(lsum, red);
  const float invden = 1.0f / denom;

  // sparsemax threshold via Michelot's algorithm (exact, sort-free):
  // tau_{k+1} = (sum_{z>tau_k} z - 1) / |{z>tau_k}| ; stop when support stable.
  float tau = -__builtin_inff();
  float nprev = -1.0f;
  for (int iter = 0; iter < 64; ++iter) {
    float s = 0.0f, c = 0.0f;
    for (int h = tid; h < H_; h += 256) {
      const float z = ldsZ[h];
      if (z > tau) { s += z; c += 1.0f; }
    }
    const float S = block_reduce_sum(s, red);
    const float C = block_reduce_sum(c, red);
    tau = (S - 1.0f) / C;
    if (C == nprev) break;   // uniform across block
    nprev = C;
  }

  // weights = 0.5*max(z - tau, 0) + 0.5*softmax(z); masked (z=-inf) -> 0 both terms
  for (int h = tid; h < H_; h += 256) {
    const float z = ldsZ[h];
    const float sp = fmaxf(z - tau, 0.0f);
    const float so = expf(z - zmax) * invden;
    zr[h] = 0.5f * sp + 0.5f * so;
  }
}

// ---------------- kernel 3: partial context, V staged via async global->LDS ----------------
// grid (NCHUNK, B), block 256. part[(b*NCHUNK+chunk)*64 + d] = sum over 128 rows.
__global__ __launch_bounds__(256)
void k3_ctx_partial(const float* __restrict__ V, const float* __restrict__ wts,
                    float* __restrict__ part)
{
  __shared__ float ldsW[ROWS_PER_BLOCK];
  __shared__ float red[256];
#if __has_builtin(__builtin_amdgcn_global_load_async_to_lds_b128)
  __shared__ float ldsV[ROWS_PER_BLOCK * D_];   // 32 KB tile
#endif

  const int b = blockIdx.y, chunk = blockIdx.x, tid = threadIdx.x;
  const size_t rowbase = (size_t)(b * H_ + chunk * ROWS_PER_BLOCK);

  if (tid < ROWS_PER_BLOCK) ldsW[tid] = wts[rowbase + tid];

#if __has_builtin(__builtin_amdgcn_global_load_async_to_lds_b128)
  {
    const float* gsrc = V + rowbase * D_;     // 128*64 floats = 2048 x b128
    #pragma unroll
    for (int i = 0; i < 8; ++i) {
      const int idx = tid + i * 256;          // float4 index
      __builtin_amdgcn_global_load_async_to_lds_b128(
          (v4i*)(gsrc + idx * 4),
          (v4i*)(ldsV + idx * 4),
          0, 0);
    }
#if __has_builtin(__builtin_amdgcn_s_wait_asynccnt)
    __builtin_amdgcn_s_wait_asynccnt(0);
#else
    asm volatile("s_wait_asynccnt 0" ::: "memory");
#endif
  }
  __syncthreads();

  const int d = tid & 63, g = tid >> 6;
  float acc = 0.0f;
  for (int hh = g; hh < ROWS_PER_BLOCK; hh += 4)
    acc += ldsW[hh] * ldsV[hh * D_ + d];
#else
  __syncthreads();
  const int d = tid & 63, g = tid >> 6;
  const float* vrow = V + rowbase * D_ + d;
  float acc = 0.0f;
  for (int hh = g; hh < ROWS_PER_BLOCK; hh += 4)
    acc += ldsW[hh] * vrow[(size_t)hh * D_];
#endif

  red[tid] = acc;
  __syncthreads();
  if (tid < D_) {
    const float s = red[tid] + red[tid + 64] + red[tid + 128] + red[tid + 192];
    part[((size_t)(b * NCHUNK + chunk)) * D_ + tid] = s;
  }
}

// ---------------- kernel 4: deterministic final reduce over chunks ----------------
__global__ __launch_bounds__(64)
void k4_reduce(const float* __restrict__ part, float* __restrict__ ctx)
{
  const int b = blockIdx.x, d = threadIdx.x;   // 64 threads
  const float* p = part + (size_t)b * NCHUNK * D_ + d;
  float s = 0.0f;
  for (int c = 0; c < NCHUNK; ++c) s += p[(size_t)c * D_];
  ctx[b * D_ + d] = s;
}

extern "C" void kernel_launch(void* const* d_in, const int* in_sizes, int n_in,
                              void* d_out, int out_size, void* d_ws, size_t ws_size,
                              hipStream_t stream) {
  const float* Q       = (const float*)d_in[0];   // [32,64]
  const float* K       = (const float*)d_in[1];   // [32,8192,64]
  const float* V       = (const float*)d_in[2];   // [32,8192,64]
  const int*   mask    = (const int*)d_in[3];     // [32,8192]
  const float* eps     = (const float*)d_in[4];   // [32,8192]
  const float* w_score = (const float*)d_in[5];   // [64]

  float* ctx = (float*)d_out;                 // [32,64]   (output 0)
  float* mu  = (float*)d_out + B_ * D_;       // [32,8192] (output 1)

  float* z_ws = (float*)d_ws;                 // 32*8192 floats (z -> weights)
  float* part = z_ws + (size_t)B_ * H_;       // 32*64*64 floats (ctx partials)

  dim3 blk(256);
  dim3 g1(NCHUNK, B_);
  k1_mu<<<g1, blk, 0, stream>>>(Q, K, mask, eps, w_score, mu, z_ws);
  k2_simplex<<<dim3(B_), blk, 0, stream>>>(z_ws);
  k3_ctx_partial<<<g1, blk, 0, stream>>>(V, z_ws, part);
  k4_reduce<<<dim3(B_), dim3(64), 0, stream>>>(part, ctx);
}